// ExpertsModule_38774964748493
// MI455X (gfx1250) — compile-verified
//
#include <hip/hip_runtime.h>

// ---------------------------------------------------------------------------
// MoE expert-gather GEMM for MI455X (gfx1250, wave32, WMMA f32 16x16x4).
// out[r] = relu(x[src[r]] @ W[e] + b[e]) with rows stably sorted by expert.
// ---------------------------------------------------------------------------

#define N_TOK 8192
#define N_EXP 8
#define DIM_D 2048
#define DIM_H 2048

#define TM 128      // M tile (tokens)
#define TN 128      // N tile (H columns)
#define KB 32       // K block staged in LDS
#define KBP 36      // padded LDS stride for A (gcd(36,64)=4 -> 16 distinct banks, 16B aligned)
#define NTHREADS 256

typedef float v2f __attribute__((ext_vector_type(2)));
typedef float v8f __attribute__((ext_vector_type(8)));

// ---------------------------------------------------------------------------
// Kernel 1: stable counting sort by expert id (single block).
// Produces src[r] = source token for sorted position r, and offs[0..8].
// ---------------------------------------------------------------------------
__global__ __launch_bounds__(NTHREADS)
void moe_build_order(const int* __restrict__ idx, int* __restrict__ src,
                     int* __restrict__ offs) {
    __shared__ int cnt[NTHREADS][N_EXP];
    __shared__ int tot[N_EXP];
    __shared__ int off[N_EXP + 1];

    const int t = threadIdx.x;
    const int CHUNK = N_TOK / NTHREADS;   // 32 tokens per thread

    int lc[N_EXP];
#pragma unroll
    for (int e = 0; e < N_EXP; ++e) lc[e] = 0;
#pragma unroll 4
    for (int i = 0; i < CHUNK; ++i) {
        int e = idx[t * CHUNK + i] & (N_EXP - 1);
        lc[e]++;
    }
#pragma unroll
    for (int e = 0; e < N_EXP; ++e) cnt[t][e] = lc[e];
    __syncthreads();

    if (t < N_EXP) {                      // exclusive scan over threads, per expert
        int s = 0;
        for (int j = 0; j < NTHREADS; ++j) { int v = cnt[j][t]; cnt[j][t] = s; s += v; }
        tot[t] = s;
    }
    __syncthreads();

    if (t == 0) {                         // exclusive scan over experts
        int s = 0;
        for (int e = 0; e < N_EXP; ++e) { off[e] = s; s += tot[e]; }
        off[N_EXP] = s;
        for (int e = 0; e <= N_EXP; ++e) offs[e] = off[e];
    }
    __syncthreads();

    int run[N_EXP];
#pragma unroll
    for (int e = 0; e < N_EXP; ++e) run[e] = off[e] + cnt[t][e];
#pragma unroll 4
    for (int i = 0; i < CHUNK; ++i) {
        int n = t * CHUNK + i;
        int e = idx[n] & (N_EXP - 1);
        src[run[e]++] = n;
    }
}

// ---------------------------------------------------------------------------
// Kernel 2: grouped GEMM. grid = (H/TN, ceil(N/TM), E), 256 threads = 8 waves.
// Wave w: wave_m = w%4 (32 rows), wave_n = w/4 (64 cols) -> 2x4 accum tiles.
// ---------------------------------------------------------------------------
__global__ __launch_bounds__(NTHREADS)
void moe_grouped_gemm(const float* __restrict__ x, const float* __restrict__ W,
                      const float* __restrict__ bias,
                      const int* __restrict__ src, const int* __restrict__ offs,
                      float* __restrict__ out) {
    const int e     = blockIdx.z;
    const int off_e = offs[e];
    const int cnt   = offs[e + 1] - off_e;
    const int m0    = blockIdx.y * TM;
    if (m0 >= cnt) return;                // uniform exit: EXEC stays all-1s
    const int n0    = blockIdx.x * TN;

    __shared__ float As[TM][KBP];         // A tile, padded stride
    __shared__ float Bs[KB][TN];          // B tile (W rows are contiguous in n)
    __shared__ int   rowid[TM];

    const int tid    = threadIdx.x;
    const int lane   = tid & 31;
    const int waveId = tid >> 5;
    const int wave_m = waveId & 3;        // 0..3  -> 32-row group
    const int wave_n = waveId >> 2;       // 0..1  -> 64-col group
    const int l16    = lane & 15;
    const int hi     = lane >> 4;         // 0: K={0,1}, 1: K={2,3}

    // Gather source-row ids once (clamped; garbage rows computed, never stored).
    if (tid < TM) {
        int g = m0 + tid;
        rowid[tid] = src[off_e + (g < cnt ? g : cnt - 1)];
    }

    const float* We = W + (size_t)e * DIM_D * DIM_H;

    v8f acc[2][4];
#pragma unroll
    for (int mi = 0; mi < 2; ++mi)
#pragma unroll
        for (int ni = 0; ni < 4; ++ni) acc[mi][ni] = (v8f)0.0f;

    for (int k0 = 0; k0 < DIM_D; k0 += KB) {
        __syncthreads();
        // --- stage A: 128 rows x 32 floats (gathered), float4 per op -------
#pragma unroll
        for (int it = 0; it < (TM * (KB / 4)) / NTHREADS; ++it) {
            int i  = tid + it * NTHREADS;
            int m  = i >> 3;              // /8 quads per row
            int kq = i & 7;
            const float4 v = *(const float4*)(x + (size_t)rowid[m] * DIM_D + k0 + kq * 4);
            *(float4*)&As[m][kq * 4] = v;
        }
        // --- stage B: 32 rows x 128 floats of W[e] -------------------------
#pragma unroll
        for (int it = 0; it < (KB * (TN / 4)) / NTHREADS; ++it) {
            int i  = tid + it * NTHREADS;
            int k  = i >> 5;              // /32 quads per row
            int nq = i & 31;
            const float4 v = *(const float4*)(We + (size_t)(k0 + k) * DIM_H + n0 + nq * 4);
            *(float4*)&Bs[k][nq * 4] = v;
        }
        __syncthreads();

        // --- 8 WMMA substeps of K=4 ---------------------------------------
#pragma unroll
        for (int kk = 0; kk < KB; kk += 4) {
            const int kf = kk + hi * 2;   // this half-wave's K pair
            v2f afrag[2];
#pragma unroll
            for (int mi = 0; mi < 2; ++mi) {
                int m = wave_m * 32 + mi * 16 + l16;
                afrag[mi].x = As[m][kf];
                afrag[mi].y = As[m][kf + 1];
            }
            v2f bfrag[4];
#pragma unroll
            for (int ni = 0; ni < 4; ++ni) {
                int n = wave_n * 64 + ni * 16 + l16;
                bfrag[ni].x = Bs[kf][n];
                bfrag[ni].y = Bs[kf + 1][n];
            }
#pragma unroll
            for (int mi = 0; mi < 2; ++mi)
#pragma unroll
                for (int ni = 0; ni < 4; ++ni)
                    acc[mi][ni] = __builtin_amdgcn_wmma_f32_16x16x4_f32(
                        /*neg_a=*/false, afrag[mi],
                        /*neg_b=*/false, bfrag[ni],
                        /*c_mod=*/(short)0, acc[mi][ni],
                        /*reuse_a=*/false, /*reuse_b=*/false);
        }
    }

    // --- epilogue: bias + relu, scatter straight into sorted output -------
#pragma unroll
    for (int ni = 0; ni < 4; ++ni) {
        const int ncol = n0 + wave_n * 64 + ni * 16 + l16;
        const float bv = bias[e * DIM_H + ncol];
#pragma unroll
        for (int mi = 0; mi < 2; ++mi) {
#pragma unroll
            for (int i = 0; i < 8; ++i) {
                int mrow = wave_m * 32 + mi * 16 + hi * 8 + i;  // C/D layout: VGPR i -> M=i / i+8
                int g = m0 + mrow;
                if (g < cnt) {
                    float v = acc[mi][ni][i] + bv;
                    out[(size_t)(off_e + g) * DIM_H + ncol] = fmaxf(v, 0.0f);
                }
            }
        }
    }
}

// ---------------------------------------------------------------------------
extern "C" void kernel_launch(void* const* d_in, const int* in_sizes, int n_in,
                              void* d_out, int out_size, void* d_ws, size_t ws_size,
                              hipStream_t stream) {
    const float* x    = (const float*)d_in[0];
    const int*   idx  = (const int*)d_in[1];
    const float* W    = (const float*)d_in[2];
    const float* bias = (const float*)d_in[3];
    float*       out  = (float*)d_out;

    int* src  = (int*)d_ws;                    // N_TOK ints
    int* offs = src + N_TOK;                   // N_EXP+1 ints

    moe_build_order<<<1, NTHREADS, 0, stream>>>(idx, src, offs);

    dim3 grid(DIM_H / TN, (N_TOK + TM - 1) / TM, N_EXP);
    moe_grouped_gemm<<<grid, NTHREADS, 0, stream>>>(x, W, bias, src, offs, out);
}